// GNN_30374008717981
// MI455X (gfx1250) — compile-verified
//
#include <hip/hip_runtime.h>
#include <math.h>

// ---------------- problem constants (from reference) ----------------
#define G_    2
#define NPN   332      // nodes per graph before pooling
#define NN    664      // total nodes
#define EE    21248    // total edges (NN * 32)
#define KH_   6        // edge-MLP bottleneck
#define K1_   166
#define K2_   83
#define DIM   332      // INDIM == D1 == D2 == R
#define D3_   128
#define KBIG  2324     // (KH+1)*332  -- fused GEMM K dimension
#define NPAD  352      // 332 padded to 32 (for 32x32 wave tiles)
#define MP1   672      // 664 padded to 32
#define MP2   352      // 332 padded to 32
#define ZDIM  1328     // (D1+D2)*2
#define EPS_  1e-5f

#define CDIV(a,b) (((a)+(b)-1)/(b))

typedef __attribute__((ext_vector_type(2))) float v2f;
typedef __attribute__((ext_vector_type(8))) float v8f;

// order-preserving float<->uint encoding for atomicMax on floats
__device__ __forceinline__ unsigned fenc(float f) {
  unsigned u = __float_as_uint(f);
  return (u & 0x80000000u) ? ~u : (u | 0x80000000u);
}
__device__ __forceinline__ float fdec(unsigned e) {
  unsigned u = (e & 0x80000000u) ? (e & 0x7fffffffu) : ~e;
  return __uint_as_float(u);
}

// ---------------- h = relu(pos @ W1), optionally gathered through perm ------
__global__ void k_hmat(const float* __restrict__ pos, const float* __restrict__ W1,
                       float* __restrict__ h, int nNodes, const int* __restrict__ perm) {
  int t = blockIdx.x * blockDim.x + threadIdx.x;
  if (t >= nNodes * KH_) return;
  int n = t / KH_, k = t % KH_;
  int row = perm ? perm[n] : n;
  const float* pr = pos + (size_t)row * DIM;
  float acc = 0.f;
  for (int i = 0; i < DIM; ++i) acc += pr[i] * W1[i * KH_ + k];
  h[t] = acc > 0.f ? acc : 0.f;
}

// ---- A'[n, k*332+i] = (k<KH ? h[n,k] : 1) * x[n,i]; zero-pad extra rows ----
__global__ void k_packA(const float* __restrict__ x, int ldx, const float* __restrict__ h,
                        float* __restrict__ A, int rows, int rowsPad) {
  int t = blockIdx.x * blockDim.x + threadIdx.x;
  if (t >= rowsPad * KBIG) return;
  int r = t / KBIG, q = t % KBIG;
  float v = 0.f;
  if (r < rows) {
    int k = q / DIM, i = q % DIM;
    float xv = x[(size_t)r * ldx + i];
    v = (k < KH_) ? h[r * KH_ + k] * xv : xv;
  }
  A[t] = v;
}

// ---- Bp pair-interleaved: Bp[(p*NPAD + n)*2 + (k&1)], p = k/2 -------------
// source row k of B' is W2 flat as (1992,332) for k<1992, else nb as (332,332)
__global__ void k_packB(const float* __restrict__ W2, const float* __restrict__ nb,
                        float* __restrict__ Bp) {
  int t = blockIdx.x * blockDim.x + threadIdx.x;
  if (t >= KBIG * NPAD) return;
  int j = t & 1;
  int pn = t >> 1;
  int p = pn / NPAD, n = pn % NPAD;
  int k = 2 * p + j;
  float v = 0.f;
  if (n < DIM)
    v = (k < KH_ * DIM) ? W2[(size_t)k * DIM + n]
                        : nb[(size_t)(k - KH_ * DIM) * DIM + n];
  Bp[t] = v;
}

// ---------------- fp32 WMMA GEMM: 32x32 C tile per wave32 -------------------
// 4 independent V_WMMA_F32_16X16X4_F32 per K-step; A and B fragments are each
// a single global_load_b64 (B thanks to pair-interleaved packing).
__global__ void k_gemm(const float* __restrict__ A, const float* __restrict__ Bp,
                       float* __restrict__ C, int Mt, int Nt) {
  int wave = blockIdx.x * (blockDim.x >> 5) + (threadIdx.x >> 5);
  int lane = threadIdx.x & 31;
  if (wave >= Mt * Nt) return;          // wave-uniform: EXEC stays all-ones
  int tm = (wave / Nt) << 5;
  int tn = (wave % Nt) << 5;
  int half = lane >> 4;                 // 0: K pair (0,1)  1: K pair (2,3)
  int sub  = lane & 15;
  int koff = half << 1;
  const float* Ar0 = A + (size_t)(tm + sub) * KBIG + koff;
  const float* Ar1 = Ar0 + (size_t)16 * KBIG;
  const float* Bc0 = Bp + ((size_t)half * NPAD + (tn + sub)) * 2;
  const float* Bc1 = Bc0 + 32;          // +16 columns, pair-interleaved
  v8f a00 = {}, a01 = {}, a10 = {}, a11 = {};
  for (int k0 = 0; k0 < KBIG; k0 += 4) {
    v2f a0, a1, b0, b1;
    a0.x = Ar0[k0];
    a0.y = Ar0[k0 + 1];
    a1.x = Ar1[k0];
    a1.y = Ar1[k0 + 1];
    size_t bo = (size_t)(k0 >> 1) * (2 * NPAD);
    b0.x = Bc0[bo];
    b0.y = Bc0[bo + 1];
    b1.x = Bc1[bo];
    b1.y = Bc1[bo + 1];
    a00 = __builtin_amdgcn_wmma_f32_16x16x4_f32(false, a0, false, b0, (short)0, a00,
                                                false, false);
    a01 = __builtin_amdgcn_wmma_f32_16x16x4_f32(false, a0, false, b1, (short)0, a01,
                                                false, false);
    a10 = __builtin_amdgcn_wmma_f32_16x16x4_f32(false, a1, false, b0, (short)0, a10,
                                                false, false);
    a11 = __builtin_amdgcn_wmma_f32_16x16x4_f32(false, a1, false, b1, (short)0, a11,
                                                false, false);
  }
  int rbase = tm + (half << 3);         // C VGPR r -> rows r / r+8 per half
  int col = tn + sub;
  float* C0 = C + (size_t)rbase * NPAD + col;
#pragma unroll
  for (int r = 0; r < 8; ++r) {
    C0[(size_t)r * NPAD] = a00[r];
    C0[(size_t)r * NPAD + 16] = a01[r];
    C0[(size_t)(r + 16) * NPAD] = a10[r];
    C0[(size_t)(r + 16) * NPAD + 16] = a11[r];
  }
}

// ---------------- edge-softmax message passing ------------------------------
__global__ void k_mp_init(unsigned* __restrict__ mx, int n) {
  int t = blockIdx.x * blockDim.x + threadIdx.x;
  if (t < n) mx[t] = fenc(1.0f);        // self-loop logit = 1.0 always present
}
__global__ void k_mp_max(const int* __restrict__ src, const int* __restrict__ dst,
                         const float* __restrict__ attr, const int* __restrict__ mask,
                         unsigned* __restrict__ mx, int nE) {
  int e = blockIdx.x * blockDim.x + threadIdx.x;
  if (e >= nE) return;
  if (mask && !mask[e]) return;
  atomicMax(&mx[dst[e]], fenc(attr[e]));
}
__global__ void k_mp_suminit(const unsigned* __restrict__ mx, float* __restrict__ sum, int n) {
  int t = blockIdx.x * blockDim.x + threadIdx.x;
  if (t < n) sum[t] = expf(1.0f - fdec(mx[t]));
}
__global__ void k_mp_sum(const int* __restrict__ src, const int* __restrict__ dst,
                         const float* __restrict__ attr, const int* __restrict__ mask,
                         const unsigned* __restrict__ mx, float* __restrict__ sum, int nE) {
  int e = blockIdx.x * blockDim.x + threadIdx.x;
  if (e >= nE) return;
  if (mask && !mask[e]) return;
  int d = dst[e];
  atomicAdd(&sum[d], expf(attr[e] - fdec(mx[d])));
}
// out[n,o] = bias[o] + alpha_self * xt[n,o]  (also zeroes the pad columns)
__global__ void k_selfbias(const float* __restrict__ xt, const float* __restrict__ bias,
                           const unsigned* __restrict__ mx, const float* __restrict__ sum,
                           float* __restrict__ out, int n) {
  int t = blockIdx.x * blockDim.x + threadIdx.x;
  if (t >= n * NPAD) return;
  int nd = t / NPAD, o = t % NPAD;
  float v = 0.f;
  if (o < DIM) {
    float aS = expf(1.0f - fdec(mx[nd])) / sum[nd];
    v = bias[o] + aS * xt[t];
  }
  out[t] = v;
}
// one wave per edge accumulates alpha * xt[src,:] into out[dst,:]
__global__ void k_mp_edges(const int* __restrict__ src, const int* __restrict__ dst,
                           const float* __restrict__ attr, const int* __restrict__ mask,
                           const unsigned* __restrict__ mx, const float* __restrict__ sum,
                           const float* __restrict__ xt, float* __restrict__ out, int nE) {
  int wave = (blockIdx.x * blockDim.x + threadIdx.x) >> 5;
  int lane = threadIdx.x & 31;
  if (wave >= nE) return;
  if (mask && !mask[wave]) return;
  int s = src[wave], d = dst[wave];
  float alpha = expf(attr[wave] - fdec(mx[d])) / sum[d];
  const float* xs = xt + (size_t)s * NPAD;
  float* od = out + (size_t)d * NPAD;
  for (int o = lane; o < DIM; o += 32) atomicAdd(&od[o], alpha * xs[o]);
}

// ---------------- TopK pooling ----------------------------------------------
__global__ void k_norm(const float* __restrict__ w, int n, float* __restrict__ outp) {
  __shared__ float red[256];
  float a = 0.f;
  for (int i = threadIdx.x; i < n; i += blockDim.x) a += w[i] * w[i];
  red[threadIdx.x] = a;
  __syncthreads();
  for (int s = blockDim.x >> 1; s > 0; s >>= 1) {
    if (threadIdx.x < s) red[threadIdx.x] += red[threadIdx.x + s];
    __syncthreads();
  }
  if (threadIdx.x == 0) *outp = sqrtf(red[0]);
}
__global__ void k_score(const float* __restrict__ x, int ld, const float* __restrict__ w,
                        const float* __restrict__ normp, float* __restrict__ score, int n) {
  int t = blockIdx.x * blockDim.x + threadIdx.x;
  if (t >= n) return;
  float s = 0.f;
  const float* xr = x + (size_t)t * ld;
  for (int i = 0; i < DIM; ++i) s += xr[i] * w[i];
  s /= *normp;
  score[t] = 1.f / (1.f + expf(-s));
}
// exact top-k by rank (ties: lower index first — matches jax.lax.top_k)
__global__ void k_select(const float* __restrict__ score, int nPer, int k,
                         int* __restrict__ perm, int* __restrict__ nidx, int n) {
  int t = blockIdx.x * blockDim.x + threadIdx.x;
  if (t >= n) return;
  int g = t / nPer, base = g * nPer;
  float s = score[t];
  int rank = 0;
  for (int m = base; m < base + nPer; ++m) {
    float sm = score[m];
    rank += (sm > s) || (sm == s && m < t);
  }
  if (rank < k) {
    perm[g * k + rank] = t;
    nidx[t] = g * k + rank;
  } else {
    nidx[t] = -1;
  }
}
__global__ void k_gather(const float* __restrict__ x, int ld, const float* __restrict__ score,
                         const int* __restrict__ perm, float* __restrict__ out, int rows) {
  int t = blockIdx.x * blockDim.x + threadIdx.x;
  if (t >= rows * NPAD) return;
  int j = t / NPAD, o = t % NPAD;
  int n = perm[j];
  out[t] = (o < DIM) ? x[(size_t)n * ld + o] * score[n] : 0.f;
}
__global__ void k_remap(const int* __restrict__ src, const int* __restrict__ dst,
                        const int* __restrict__ nidx, int* __restrict__ src2,
                        int* __restrict__ dst2, int* __restrict__ msk2, int nE) {
  int e = blockIdx.x * blockDim.x + threadIdx.x;
  if (e >= nE) return;
  int ns = nidx[src[e]], nd = nidx[dst[e]];
  int m = (ns >= 0) && (nd >= 0);
  src2[e] = m ? ns : 0;
  dst2[e] = m ? nd : 0;
  msk2[e] = m;
}

// ---------------- readout / head --------------------------------------------
__global__ void k_readout(const float* __restrict__ x, int ld, int k,
                          float* __restrict__ z, int zoff) {
  int t = blockIdx.x * blockDim.x + threadIdx.x;
  if (t >= G_ * DIM) return;
  int g = t / DIM, c = t % DIM;
  float mx = -INFINITY, sm = 0.f;
  for (int j = 0; j < k; ++j) {
    float v = x[(size_t)(g * k + j) * ld + c];
    mx = fmaxf(mx, v);
    sm += v;
  }
  z[g * ZDIM + zoff + c] = mx;
  z[g * ZDIM + zoff + DIM + c] = sm / (float)k;
}
__global__ void k_fc(const float* __restrict__ in, int inDim, const float* __restrict__ W,
                     const float* __restrict__ bias, float* __restrict__ out, int outDim) {
  int t = blockIdx.x * blockDim.x + threadIdx.x;
  if (t >= G_ * outDim) return;
  int g = t / outDim, c = t % outDim;
  float acc = bias[c];
  const float* ir = in + (size_t)g * inDim;
  for (int j = 0; j < inDim; ++j) acc += ir[j] * W[(size_t)j * outDim + c];
  out[t] = acc;
}
__global__ void k_bnrelu(float* __restrict__ tb, int dim, const float* __restrict__ gm,
                         const float* __restrict__ bt) {
  int t = blockIdx.x * blockDim.x + threadIdx.x;
  if (t >= dim) return;
  float a = tb[t], b = tb[dim + t];
  float m = 0.5f * (a + b);
  float da = a - m, db = b - m;
  float var = 0.5f * (da * da + db * db);
  float inv = rsqrtf(var + EPS_);
  float r0 = da * inv * gm[t] + bt[t];
  float r1 = db * inv * gm[t] + bt[t];
  tb[t] = r0 > 0.f ? r0 : 0.f;
  tb[dim + t] = r1 > 0.f ? r1 : 0.f;
}

// ---------------- driver ----------------------------------------------------
extern "C" void kernel_launch(void* const* d_in, const int* in_sizes, int n_in,
                              void* d_out, int out_size, void* d_ws, size_t ws_size,
                              hipStream_t stream) {
  (void)in_sizes; (void)n_in; (void)out_size; (void)ws_size;
  const float* x     = (const float*)d_in[0];
  const int*   ei    = (const int*)d_in[1];
  const float* eattr = (const float*)d_in[3];
  const float* pos   = (const float*)d_in[4];
  const float* c1W1  = (const float*)d_in[6];
  const float* c1W2  = (const float*)d_in[7];
  const float* c1nb  = (const float*)d_in[8];
  const float* c1b   = (const float*)d_in[9];
  const float* p1w   = (const float*)d_in[10];
  const float* c2W1  = (const float*)d_in[11];
  const float* c2W2  = (const float*)d_in[12];
  const float* c2nb  = (const float*)d_in[13];
  const float* c2b   = (const float*)d_in[14];
  const float* p2w   = (const float*)d_in[15];
  const float* fc1W  = (const float*)d_in[16];
  const float* fc1b  = (const float*)d_in[17];
  const float* bn1g  = (const float*)d_in[18];
  const float* bn1b  = (const float*)d_in[19];
  const float* fc2W  = (const float*)d_in[20];
  const float* fc2b  = (const float*)d_in[21];
  const float* bn2g  = (const float*)d_in[22];
  const float* bn2b  = (const float*)d_in[23];
  const float* fc3W  = (const float*)d_in[24];
  const float* fc3b  = (const float*)d_in[25];
  const int* src = ei;
  const int* dst = ei + EE;

  char* cur = (char*)d_ws;
  auto alloc = [&](size_t bytes) -> void* {
    void* p = (void*)cur;
    cur += (bytes + 255) & ~(size_t)255;
    return p;
  };
  float*    h1   = (float*)alloc((size_t)NN * KH_ * 4);
  float*    h2   = (float*)alloc((size_t)332 * KH_ * 4);
  float*    Abuf = (float*)alloc((size_t)MP1 * KBIG * 4);   // reused conv2
  float*    Bbuf = (float*)alloc((size_t)KBIG * NPAD * 4);  // reused conv2
  float*    xtb  = (float*)alloc((size_t)MP1 * NPAD * 4);   // reused conv2
  float*    out1 = (float*)alloc((size_t)NN * NPAD * 4);
  float*    out2 = (float*)alloc((size_t)332 * NPAD * 4);
  float*    x1p  = (float*)alloc((size_t)332 * NPAD * 4);
  float*    x2p  = (float*)alloc((size_t)166 * NPAD * 4);
  unsigned* mxe  = (unsigned*)alloc((size_t)NN * 4);
  float*    sume = (float*)alloc((size_t)NN * 4);
  float*    sc1  = (float*)alloc((size_t)NN * 4);
  float*    sc2  = (float*)alloc((size_t)332 * 4);
  int*      prm1 = (int*)alloc((size_t)332 * 4);
  int*      prm2 = (int*)alloc((size_t)166 * 4);
  int*      nidx = (int*)alloc((size_t)NN * 4);
  int*      src2 = (int*)alloc((size_t)EE * 4);
  int*      dst2 = (int*)alloc((size_t)EE * 4);
  int*      msk2 = (int*)alloc((size_t)EE * 4);
  float*    nrm  = (float*)alloc(64);
  float*    zb   = (float*)alloc((size_t)G_ * ZDIM * 4);
  float*    t1   = (float*)alloc((size_t)G_ * 332 * 4);
  float*    t2   = (float*)alloc((size_t)G_ * D3_ * 4);

  const dim3 B(256);
  // ---------------- conv1 (fused GEMM via fp32 WMMA, 32x32/wave) ----------
  k_hmat<<<CDIV(NN * KH_, 256), B, 0, stream>>>(pos, c1W1, h1, NN, nullptr);
  k_packA<<<CDIV(MP1 * KBIG, 256), B, 0, stream>>>(x, DIM, h1, Abuf, NN, MP1);
  k_packB<<<CDIV(KBIG * NPAD, 256), B, 0, stream>>>(c1W2, c1nb, Bbuf);
  k_gemm<<<CDIV((MP1 / 32) * (NPAD / 32), 8), B, 0, stream>>>(Abuf, Bbuf, xtb,
                                                              MP1 / 32, NPAD / 32);
  k_mp_init<<<CDIV(NN, 256), B, 0, stream>>>(mxe, NN);
  k_mp_max<<<CDIV(EE, 256), B, 0, stream>>>(src, dst, eattr, nullptr, mxe, EE);
  k_mp_suminit<<<CDIV(NN, 256), B, 0, stream>>>(mxe, sume, NN);
  k_mp_sum<<<CDIV(EE, 256), B, 0, stream>>>(src, dst, eattr, nullptr, mxe, sume, EE);
  k_selfbias<<<CDIV(NN * NPAD, 256), B, 0, stream>>>(xtb, c1b, mxe, sume, out1, NN);
  k_mp_edges<<<CDIV(EE * 32, 256), B, 0, stream>>>(src, dst, eattr, nullptr, mxe, sume,
                                                   xtb, out1, EE);
  // ---------------- pool1 + readout1 ----------------
  k_norm<<<1, B, 0, stream>>>(p1w, DIM, nrm);
  k_score<<<CDIV(NN, 256), B, 0, stream>>>(out1, NPAD, p1w, nrm, sc1, NN);
  k_select<<<CDIV(NN, 256), B, 0, stream>>>(sc1, NPN, K1_, prm1, nidx, NN);
  k_gather<<<CDIV(332 * NPAD, 256), B, 0, stream>>>(out1, NPAD, sc1, prm1, x1p, 332);
  k_remap<<<CDIV(EE, 256), B, 0, stream>>>(src, dst, nidx, src2, dst2, msk2, EE);
  k_readout<<<CDIV(G_ * DIM, 256), B, 0, stream>>>(x1p, NPAD, K1_, zb, 0);
  // ---------------- conv2 ----------------
  k_hmat<<<CDIV(332 * KH_, 256), B, 0, stream>>>(pos, c2W1, h2, 332, prm1);
  k_packA<<<CDIV(MP2 * KBIG, 256), B, 0, stream>>>(x1p, NPAD, h2, Abuf, 332, MP2);
  k_packB<<<CDIV(KBIG * NPAD, 256), B, 0, stream>>>(c2W2, c2nb, Bbuf);
  k_gemm<<<CDIV((MP2 / 32) * (NPAD / 32), 8), B, 0, stream>>>(Abuf, Bbuf, xtb,
                                                              MP2 / 32, NPAD / 32);
  k_mp_init<<<CDIV(332, 256), B, 0, stream>>>(mxe, 332);
  k_mp_max<<<CDIV(EE, 256), B, 0, stream>>>(src2, dst2, eattr, msk2, mxe, EE);
  k_mp_suminit<<<CDIV(332, 256), B, 0, stream>>>(mxe, sume, 332);
  k_mp_sum<<<CDIV(EE, 256), B, 0, stream>>>(src2, dst2, eattr, msk2, mxe, sume, EE);
  k_selfbias<<<CDIV(332 * NPAD, 256), B, 0, stream>>>(xtb, c2b, mxe, sume, out2, 332);
  k_mp_edges<<<CDIV(EE * 32, 256), B, 0, stream>>>(src2, dst2, eattr, msk2, mxe, sume,
                                                   xtb, out2, EE);
  // ---------------- pool2 + readout2 ----------------
  k_norm<<<1, B, 0, stream>>>(p2w, DIM, nrm);
  k_score<<<CDIV(332, 256), B, 0, stream>>>(out2, NPAD, p2w, nrm, sc2, 332);
  k_select<<<CDIV(332, 256), B, 0, stream>>>(sc2, K1_, K2_, prm2, nidx, 332);
  k_gather<<<CDIV(166 * NPAD, 256), B, 0, stream>>>(out2, NPAD, sc2, prm2, x2p, 166);
  k_readout<<<CDIV(G_ * DIM, 256), B, 0, stream>>>(x2p, NPAD, K2_, zb, 2 * DIM);
  // ---------------- MLP head ----------------
  k_fc<<<CDIV(G_ * DIM, 256), B, 0, stream>>>(zb, ZDIM, fc1W, fc1b, t1, DIM);
  k_bnrelu<<<CDIV(DIM, 256), B, 0, stream>>>(t1, DIM, bn1g, bn1b);
  k_fc<<<CDIV(G_ * D3_, 256), B, 0, stream>>>(t1, DIM, fc2W, fc2b, t2, D3_);
  k_bnrelu<<<CDIV(D3_, 256), B, 0, stream>>>(t2, D3_, bn2g, bn2b);
  k_fc<<<1, B, 0, stream>>>(t2, D3_, fc3W, fc3b, (float*)d_out, 2);
}